// GConvII_78529182040078
// MI455X (gfx1250) — compile-verified
//
#include <hip/hip_runtime.h>

#define N_NODES 50000
#define N_EDGES 800000
#define D       128
#define ALPHA   0.1f
#define BETA    0.5f

#define EDGES_PER_WAVE 32
#define TILE_ROWS      16
#define WAVES_PER_BLK  4
#define ROW_TILES      (N_NODES / TILE_ROWS)   // 3125, exact
#define LDSP           (D + 4)                 // pad 4 floats -> conflict-free ds_load_b64

typedef float v2f __attribute__((ext_vector_type(2)));
typedef float v8f __attribute__((ext_vector_type(8)));

// ---------------------------------------------------------------------------
// Kernel 1: zero the hi accumulator in workspace (ws is poisoned, re-zero per call)
// ---------------------------------------------------------------------------
__global__ __launch_bounds__(256)
void gcn_zero(float4* __restrict__ p, int n4) {
    int i = blockIdx.x * blockDim.x + threadIdx.x;
    if (i < n4) {
        float4 z; z.x = 0.f; z.y = 0.f; z.z = 0.f; z.w = 0.f;
        p[i] = z;
    }
}

// ---------------------------------------------------------------------------
// Kernel 2: SpMM scatter. One wave per 32 contiguous (sorted-by-row) edges.
// Lane j owns features [4j..4j+3]. Register-accumulate runs of equal rows,
// flush with hardware fp32 global atomics only at row boundaries.
// ---------------------------------------------------------------------------
__global__ __launch_bounds__(256)
void gcn_scatter(const float* __restrict__ h,
                 const int*   __restrict__ erow,
                 const int*   __restrict__ ecol,
                 const float* __restrict__ evals,
                 float*       __restrict__ hi)
{
    const int gwave = (blockIdx.x * blockDim.x + threadIdx.x) >> 5;
    const int lane  = threadIdx.x & 31;
    const int e0    = gwave * EDGES_PER_WAVE;
    if (e0 >= N_EDGES) return;
    const int e1 = min(e0 + EDGES_PER_WAVE, N_EDGES);
    const int f  = lane * 4;

    float ax = 0.f, ay = 0.f, az = 0.f, aw = 0.f;
    int cur = erow[e0];

    for (int e = e0; e < e1; ++e) {
        const int   col = ecol[e];
        const float val = evals[e];
        const float4 s = *(const float4*)(h + (size_t)col * D + f);
        ax = fmaf(val, s.x, ax);
        ay = fmaf(val, s.y, ay);
        az = fmaf(val, s.z, az);
        aw = fmaf(val, s.w, aw);
        const int nxt = (e + 1 < e1) ? erow[e + 1] : -1;
        if (nxt != cur) {
            float* dst = hi + (size_t)cur * D + f;
            unsafeAtomicAdd(dst + 0, ax);
            unsafeAtomicAdd(dst + 1, ay);
            unsafeAtomicAdd(dst + 2, az);
            unsafeAtomicAdd(dst + 3, aw);
            ax = ay = az = aw = 0.f;
            cur = nxt;
        }
    }
}

// ---------------------------------------------------------------------------
// Kernel 3: fused support + (support @ W.T + b) + output blend, fp32 WMMA.
// Block = 128 threads = 4 waves. Block stages a 64x128 support panel in LDS;
// each wave computes one 16-row tile against all 8 column tiles of W^T with
// v_wmma_f32_16x16x4_f32 (K=4 per step, 32 steps).
// ---------------------------------------------------------------------------
__global__ __launch_bounds__(128)
void gcn_gemm(const float* __restrict__ hi,
              const float* __restrict__ h0,
              const float* __restrict__ W,
              const float* __restrict__ bias,
              const int*   __restrict__ lptr,
              float*       __restrict__ out)
{
    __shared__ float sp[64 * LDSP];   // 64 rows x (128+4) floats = 33792 B

    const int lane  = threadIdx.x & 31;
    const int wv    = threadIdx.x >> 5;
    const int tile0 = blockIdx.x * WAVES_PER_BLK;
    const int rbase = tile0 * TILE_ROWS;

    const float theta = logf(BETA / (float)(*lptr) + 1.0f);
    const float omt   = 1.0f - theta;

    // Cooperative staging: support = (1-ALPHA)*hi + ALPHA*h0 into LDS.
    const int nrows = min(64, N_NODES - rbase);
    for (int i = threadIdx.x; i < nrows * (D / 4); i += 128) {
        const int r = i >> 5;            // 0..63
        const int c = (i & 31) * 4;      // 0..124
        const size_t g = (size_t)(rbase + r) * D + c;
        const float4 a  = *(const float4*)(hi + g);
        const float4 b0 = *(const float4*)(h0 + g);
        float* s = sp + r * LDSP + c;
        s[0] = (1.0f - ALPHA) * a.x + ALPHA * b0.x;
        s[1] = (1.0f - ALPHA) * a.y + ALPHA * b0.y;
        s[2] = (1.0f - ALPHA) * a.z + ALPHA * b0.z;
        s[3] = (1.0f - ALPHA) * a.w + ALPHA * b0.w;
    }
    __syncthreads();

    const int myTile = tile0 + wv;
    if (myTile >= ROW_TILES) return;     // wave-uniform: EXEC all-ones at WMMA

    // A fragment (16x4 f32): lanes 0-15 -> row=lane, K=k,k+1 ; lanes 16-31 -> K=k+2,k+3
    const int am   = lane & 15;
    const int koff = (lane >> 4) * 2;
    const float* arow = sp + (wv * TILE_ROWS + am) * LDSP;

    for (int ct = 0; ct < 8; ++ct) {
        const int c0  = ct * 16;
        const int n16 = lane & 15;
        // B fragment (4x16 f32) of W^T: B[k][n] = W[n][k]
        const float* brow = W + (size_t)(c0 + n16) * D;

        v8f acc = {};
        #pragma unroll 8
        for (int k = 0; k < D; k += 4) {
            v2f a = *(const v2f*)(arow + k + koff);   // LDS, 8B aligned
            v2f b = *(const v2f*)(brow + k + koff);   // global, cache-resident
            acc = __builtin_amdgcn_wmma_f32_16x16x4_f32(
                      false, a, false, b, (short)0, acc, false, false);
        }

        // Epilogue: D layout lanes 0-15 -> M=v, lanes 16-31 -> M=v+8, N=lane&15
        const float bb = bias[c0 + n16];
        const int   mb = (lane >> 4) * 8;
        #pragma unroll
        for (int v = 0; v < 8; ++v) {
            const int m   = mb + v;
            const float sup = sp[(wv * TILE_ROWS + m) * LDSP + c0 + n16];
            out[(size_t)(rbase + wv * TILE_ROWS + m) * D + c0 + n16] =
                theta * (acc[v] + bb) + omt * sup;
        }
    }
}

// ---------------------------------------------------------------------------
extern "C" void kernel_launch(void* const* d_in, const int* in_sizes, int n_in,
                              void* d_out, int out_size, void* d_ws, size_t ws_size,
                              hipStream_t stream) {
    const float* h    = (const float*)d_in[0];
    const float* h0   = (const float*)d_in[1];
    const int*   erow = (const int*)  d_in[2];
    const int*   ecol = (const int*)  d_in[3];
    const float* ev   = (const float*)d_in[4];
    const float* W    = (const float*)d_in[5];
    const float* bias = (const float*)d_in[6];
    const int*   l    = (const int*)  d_in[7];
    float* out = (float*)d_out;
    float* hi  = (float*)d_ws;   // N_NODES*D fp32 accumulator (25.6 MB)

    const int n4 = (N_NODES * D) / 4;
    gcn_zero<<<(n4 + 255) / 256, 256, 0, stream>>>((float4*)d_ws, n4);

    const int waves   = (N_EDGES + EDGES_PER_WAVE - 1) / EDGES_PER_WAVE;
    const int threads = waves * 32;
    gcn_scatter<<<(threads + 255) / 256, 256, 0, stream>>>(h, erow, ecol, ev, hi);

    gcn_gemm<<<(ROW_TILES + WAVES_PER_BLK - 1) / WAVES_PER_BLK, 128, 0, stream>>>(
        hi, h0, W, bias, l, out);
}